// FSCosNegBBoxHead1_60928406061675
// MI455X (gfx1250) — compile-verified
//
#include <hip/hip_runtime.h>
#include <math.h>

// ---------------------------------------------------------------------------
// FSCosNegBBoxHead — MI455X (gfx1250, wave32, WMMA)
//
// Dominant cost: reading x (822 MB) -> ~35us HBM floor. Everything else is
// fused small GEMMs done with V_WMMA_F32_16X16X4_F32 (fp32 for numerical
// fidelity through inverse_sigmoid). Kernel A stages x tiles into LDS with
// the CDNA5 async Global->LDS path (ASYNCcnt) when the builtin is available.
// ---------------------------------------------------------------------------

typedef __attribute__((ext_vector_type(2))) float v2f;
typedef __attribute__((ext_vector_type(8))) float v8f;
typedef int v4i_ __attribute__((vector_size(16)));   // matches builtin param type

#define BSZ        16384
#define CH         256
#define RR         49          // 7*7
#define NKEEP      60          // cls_score[:, :60]
#define NBBOX      244         // (60+1)*4 bbox columns kept
#define NCLS_COLS  240         // 60 groups of [cos, neg0, neg1, neg2]
#define NCOLS      496         // 240 cls + 256 (244 bbox + 12 zero pad)
#define NTILES     31          // 496 / 16
#define EPS_NORM_  1e-12f
#define EPS_IS_    1e-5f
#define COS_SCALE_ 3.0f
#define NEG_BETA_  0.3f
#define LDSTR      260         // padded LDS row stride (dwords): conflict-free b64 frags

#if defined(__has_builtin)
#if __has_builtin(__builtin_amdgcn_global_load_async_to_lds_b128)
#define HAVE_ASYNC_LDS 1
#endif
#if __has_builtin(__builtin_amdgcn_s_wait_asynccnt)
#define HAVE_WAIT_ASYNC 1
#endif
#endif

__device__ __forceinline__ float blockReduceSum256(float v, float* sred) {
  #pragma unroll
  for (int off = 16; off > 0; off >>= 1) v += __shfl_down(v, off);
  int wid = threadIdx.x >> 5, lid = threadIdx.x & 31;
  if (lid == 0) sred[wid] = v;
  __syncthreads();
  if (threadIdx.x == 0) {
    float tot = 0.f;
    #pragma unroll
    for (int i = 0; i < 8; ++i) tot += sred[i];
    sred[0] = tot;
  }
  __syncthreads();
  return sred[0];
}

// ---- Kernel A: feat = mean(x, (2,3)); feat_norm = l2norm(feat) -------------
// One block per batch element. Stage 49KB in LDS; prefer CDNA5 async
// Global->LDS b128 DMA (no VGPR round-trip), tracked with ASYNCcnt.
__global__ __launch_bounds__(256) void k_feat(const float* __restrict__ x,
                                              float* __restrict__ feat,
                                              float* __restrict__ featn) {
  __shared__ float sx[CH * RR];          // 12544 floats = 49KB
  __shared__ float sred[8];
  const int b = blockIdx.x;
  const int t = threadIdx.x;
  const float4* x4 = (const float4*)(x + (size_t)b * (CH * RR)); // 16B aligned
  float4* s4 = (float4*)sx;

#if defined(HAVE_ASYNC_LDS)
  {
    #pragma unroll
    for (int i = 0; i < 12; ++i)
      __builtin_amdgcn_global_load_async_to_lds_b128(
          (v4i_*)(x4 + i * 256 + t), (v4i_*)(s4 + i * 256 + t), 0, 0);
    if (t < 64)
      __builtin_amdgcn_global_load_async_to_lds_b128(
          (v4i_*)(x4 + 3072 + t), (v4i_*)(s4 + 3072 + t), 0, 0);
#if defined(HAVE_WAIT_ASYNC)
    __builtin_amdgcn_s_wait_asynccnt(0);
#else
    asm volatile("s_wait_asynccnt 0x0" ::: "memory");
#endif
  }
#else
  #pragma unroll
  for (int i = 0; i < 12; ++i) s4[i * 256 + t] = x4[i * 256 + t];
  if (t < 64) s4[3072 + t] = x4[3072 + t];   // 3136 float4 total
#endif
  __syncthreads();

  float sum = 0.f;
  const int base = t * RR;                   // stride-49 across lanes: bank-clean
  #pragma unroll
  for (int j = 0; j < RR; ++j) sum += sx[base + j];
  const float f = sum * (1.0f / 49.0f);

  const float ss = blockReduceSum256(f * f, sred);
  const float fn = f / fmaxf(sqrtf(ss), EPS_NORM_);
  feat [(size_t)b * CH + t] = f;
  featn[(size_t)b * CH + t] = fn;
}

// ---- Kernel B1: fg_w_norm rows 0..59; also write cls group col 4k ----------
__global__ __launch_bounds__(256) void k_fgn(const float* __restrict__ W_cls,
                                             float* __restrict__ fgn,
                                             float* __restrict__ Wc) {
  __shared__ float sred[8];
  const int k = blockIdx.x, t = threadIdx.x;
  const float w  = W_cls[(size_t)k * CH + t];
  const float ss = blockReduceSum256(w * w, sred);
  const float v  = w / fmaxf(sqrtf(ss), EPS_NORM_);
  fgn[(size_t)k * CH + t] = v;
  Wc[(size_t)(4 * k) * CH + t] = v;          // group slot 0 = cos weight
}

// ---- Kernel B2: neg_reps_norm[k][m] -> Wc col 4k+1+m -----------------------
__global__ __launch_bounds__(256) void k_negrep(const float* __restrict__ W_no,
                                                const float* __restrict__ b_no,
                                                const float* __restrict__ fgn,
                                                float* __restrict__ Wc) {
  __shared__ float sF[CH];
  __shared__ float sred[8];
  const int k = blockIdx.x, m = blockIdx.y, d = threadIdx.x;
  sF[d] = fgn[(size_t)k * CH + d];
  __syncthreads();
  const int j = m * CH + d;
  const float* wrow = W_no + (size_t)j * CH;
  float dot = 0.f;
  #pragma unroll 4
  for (int dd = 0; dd < CH; ++dd) dot += sF[dd] * wrow[dd];
  const float rep = sF[d] + dot + b_no[j];
  const float ss  = blockReduceSum256(rep * rep, sred);
  const float v   = rep / fmaxf(sqrtf(ss), EPS_NORM_);
  Wc[(size_t)(4 * k + 1 + m) * CH + d] = v;
}

// ---- Kernel B3: W_reg rows 0..243 -> Wc cols 240..483; pad 484..495 --------
__global__ __launch_bounds__(256) void k_wreg(const float* __restrict__ W_reg,
                                              float* __restrict__ Wc) {
  const int j = blockIdx.x;                  // 0..255
  const int d = threadIdx.x;
  const float v = (j < NBBOX) ? W_reg[(size_t)j * CH + d] : 0.f;
  Wc[(size_t)(NCLS_COLS + j) * CH + d] = v;
}

// ---- Kernel C: WMMA GEMM (M=16384, N=496, K=256) + fused epilogue ----------
// Block = 16 M-rows, 8 waves; wave handles N-tiles {w, w+8, w+16, w+24}.
// Tiles 0..14 use feat_norm (cls, grouped cols); 15..30 use feat (bbox).
__global__ __launch_bounds__(256) void k_gemm(const float* __restrict__ feat,
                                              const float* __restrict__ featn,
                                              const float* __restrict__ Wc,
                                              const float* __restrict__ b_reg,
                                              float* __restrict__ out) {
  __shared__ float sFN[16 * LDSTR];
  __shared__ float sFE[16 * LDSTR];
  const int t   = threadIdx.x;
  const int m0  = blockIdx.x * 16;
  #pragma unroll
  for (int i = 0; i < 16; ++i) {
    sFN[i * LDSTR + t] = featn[(size_t)(m0 + i) * CH + t];
    sFE[i * LDSTR + t] = feat [(size_t)(m0 + i) * CH + t];
  }
  __syncthreads();

  const int wave = t >> 5;
  const int lane = t & 31;
  const int hi   = lane >> 4;                // 0/1: which K half of the frag
  const int ln   = lane & 15;                // N (for B/D) or M (for A)

  for (int it = 0; it < 4; ++it) {
    const int tile = wave + 8 * it;
    if (tile >= NTILES) break;               // wave-uniform: EXEC stays full
    const bool isCls = tile < 15;
    const float* Acol = (isCls ? sFN : sFE) + ln * LDSTR + 2 * hi;
    const float* Bcol = Wc + ((size_t)tile * 16 + ln) * CH + 2 * hi;

    v8f acc = {};
    #pragma unroll 8
    for (int k0 = 0; k0 < CH; k0 += 4) {
      v2f a = *(const v2f*)(Acol + k0);      // A[M=ln][k0+2hi .. +1]
      v2f b = *(const v2f*)(Bcol + k0);      // B[k0+2hi .. +1][N=ln]
      acc = __builtin_amdgcn_wmma_f32_16x16x4_f32(
          false, a, false, b, (short)0, acc, false, false);
    }

    if (isCls) {
      // cols grouped per k: [cos, neg0, neg1, neg2]; lanes ln%4==0 finalize.
      #pragma unroll
      for (int v = 0; v < 8; ++v) {
        const float val = acc[v];
        const float n1 = __shfl_down(val, 1);
        const float n2 = __shfl_down(val, 2);
        const float n3 = __shfl_down(val, 3);
        if ((ln & 3) == 0) {
          const int row  = m0 + v + 8 * hi;
          const int kcls = (tile * 16 + ln) >> 2;
          const float neg = fmaxf(n1, fmaxf(n2, n3));
          const float s   = val - NEG_BETA_ * neg;
          const float d1  = 1.0f - s;
          float cls = expf(-d1 * d1 * COS_SCALE_);
          cls = fminf(fmaxf(cls, 0.0f), 1.0f);
          const float x1 = fmaxf(cls, EPS_IS_);
          const float x2 = fmaxf(1.0f - cls, EPS_IS_);
          out[(size_t)row * NKEEP + kcls] = logf(x1 / x2);
        }
      }
    } else {
      const int col = (tile - 15) * 16 + ln; // 0..255 bbox col
      if (col < NBBOX) {
        const float bias = b_reg[col];
        #pragma unroll
        for (int v = 0; v < 8; ++v) {
          const int row = m0 + v + 8 * hi;
          out[(size_t)BSZ * NKEEP + (size_t)row * NBBOX + col] = acc[v] + bias;
        }
      }
    }
  }
}

// ---------------------------------------------------------------------------
extern "C" void kernel_launch(void* const* d_in, const int* in_sizes, int n_in,
                              void* d_out, int out_size, void* d_ws, size_t ws_size,
                              hipStream_t stream) {
  (void)in_sizes; (void)n_in; (void)out_size; (void)ws_size;
  const float* x     = (const float*)d_in[0];
  const float* W_cls = (const float*)d_in[1];
  const float* W_no  = (const float*)d_in[2];
  const float* b_no  = (const float*)d_in[3];
  const float* W_reg = (const float*)d_in[4];
  const float* b_reg = (const float*)d_in[5];
  float* out = (float*)d_out;

  float* ws    = (float*)d_ws;
  float* feat  = ws;                                   // 16384*256
  float* featn = feat  + (size_t)BSZ * CH;             // 16384*256
  float* fgn   = featn + (size_t)BSZ * CH;             // 60*256
  float* Wc    = fgn   + (size_t)NKEEP * CH;           // 496*256 (col-major per col)

  k_feat  <<<BSZ,         256, 0, stream>>>(x, feat, featn);
  k_fgn   <<<NKEEP,       256, 0, stream>>>(W_cls, fgn, Wc);
  k_negrep<<<dim3(NKEEP, 3), 256, 0, stream>>>(W_no, b_no, fgn, Wc);
  k_wreg  <<<256,         256, 0, stream>>>(W_reg, Wc);
  k_gemm  <<<BSZ / 16,    256, 0, stream>>>(feat, featn, Wc, b_reg, out);
}